// DTRR_60851096649748
// MI455X (gfx1250) — compile-verified
//
#include <hip/hip_runtime.h>
#include <hip/hip_bf16.h>

// Problem constants (from reference)
#define BB   16
#define LL   512
#define HH   256
#define RR   8
#define EE   524288
#define NN   (BB * 2 * LL)        // 16384 nodes
#define KTOT (RR * HH + HH)       // 2304 = 8 relations * 256 + root block
#define KREL (RR * HH)            // 2048
#define SUMS_ELEMS ((size_t)NN * RR * HH)   // 33,554,432 floats = 128 MB
#define CNT_ELEMS  ((size_t)NN * RR)        // 131,072
#define BPACK_ELEMS ((size_t)KTOT * HH)     // 589,824 floats = 2.36 MB
#define APAD 68                             // A-panel row stride: 16B-aligned, conflict-free

typedef __attribute__((ext_vector_type(2))) float v2f;
typedef __attribute__((ext_vector_type(8))) float v8f;

// ---- helpers -------------------------------------------------------------

// Node n lives in input_s (first L rows of each batch) or input_a (last L).
__device__ __forceinline__ const float* node_row(const float* __restrict__ s,
                                                 const float* __restrict__ a,
                                                 int n) {
    int b = n >> 10;            // 2L = 1024
    int j = n & 1023;
    return (j < LL) ? (s + ((size_t)(b * LL + j)) * HH)
                    : (a + ((size_t)(b * LL + (j - LL))) * HH);
}

// Output is (sent flat, act flat) concatenated.
__device__ __forceinline__ size_t out_off(int n) {
    int b = n >> 10;
    int j = n & 1023;
    return (j < LL) ? ((size_t)(b * LL + j)) * HH
                    : (size_t)BB * LL * HH + ((size_t)(b * LL + (j - LL))) * HH;
}

static __device__ __forceinline__ v8f wmma_f32x4(v2f a, v2f b, v8f c) {
    // 8 args: (neg_a, A, neg_b, B, c_mod, C, reuse_a, reuse_b)
    return __builtin_amdgcn_wmma_f32_16x16x4_f32(false, a, false, b,
                                                 (short)0, c, false, false);
}

// Async DMA: 16 bytes global -> LDS, tracked by ASYNCcnt (no VGPR round trip).
static __device__ __forceinline__ void async_b128(unsigned lds_byte, const float* g) {
    asm volatile("global_load_async_to_lds_b128 %0, %1, off"
                 :: "v"(lds_byte), "v"(g) : "memory");
}
static __device__ __forceinline__ void wait_async0() {
    asm volatile("s_wait_asynccnt 0x0" ::: "memory");
}

// ---- kernel 1: zero the workspace (sums + counts, contiguous) ------------
__global__ __launch_bounds__(256) void k_zero(float4* __restrict__ ws, size_t n4) {
    size_t i = (size_t)blockIdx.x * blockDim.x + threadIdx.x;
    if (i < n4) ws[i] = make_float4(0.f, 0.f, 0.f, 0.f);
}

// ---- kernel 2: fused count + scatter-add ---------------------------------
// One block (256 threads) per edge: lanes cover the H=256 message row.
__global__ __launch_bounds__(256) void k_scatter(
    const float* __restrict__ in_s, const float* __restrict__ in_a,
    const long long* __restrict__ edge_index,   // (2,E)
    const long long* __restrict__ edge_type,    // (E,)
    float* __restrict__ sums, unsigned int* __restrict__ cnt) {
    int e = blockIdx.x;
    int src = (int)edge_index[e];
    int dst = (int)edge_index[(size_t)EE + e];
    int t   = (int)edge_type[e];
    int h   = threadIdx.x;
    size_t seg = (size_t)dst * RR + t;
    if (h == 0) atomicAdd(&cnt[seg], 1u);
    const float* xr = node_row(in_s, in_a, src);
    atomicAdd(&sums[seg * HH + h], xr[h]);
}

// ---- kernel 3: sums -> means (scale in place) ----------------------------
__global__ __launch_bounds__(256) void k_scale(float* __restrict__ sums,
                                               const unsigned int* __restrict__ cnt) {
    size_t i4 = (size_t)blockIdx.x * blockDim.x + threadIdx.x;  // one float4 each
    size_t i  = i4 * 4;
    if (i >= SUMS_ELEMS) return;
    unsigned int c = cnt[i >> 8];                 // H = 256 elems per segment
    float m = (c > 1u) ? (1.0f / (float)c) : 1.0f;
    float4* p = (float4*)(sums + i);
    float4 v = *p;
    v.x *= m; v.y *= m; v.z *= m; v.w *= m;
    *p = v;
}

// ---- kernel 4: pack B = [weight ; root] into [k/4][col][4] ---------------
// Bpack[(ks*HH + c)*4 + j] = B(4*ks + j, c). Lane fragments (k+off,k+off+1)
// become one contiguous, 8B-aligned float2 -> single global_load_b64.
__global__ __launch_bounds__(256) void k_prepB(const float* __restrict__ w,
                                               const float* __restrict__ r,
                                               float* __restrict__ Bpack) {
    int idx = blockIdx.x * 256 + threadIdx.x;   // over KTOT*HH elements
    int j  = idx & 3;
    int c  = (idx >> 2) & (HH - 1);
    int ks = idx >> 10;
    int gk = ks * 4 + j;
    Bpack[idx] = (gk < KREL) ? w[(size_t)gk * HH + c]
                             : r[(size_t)(gk - KREL) * HH + c];
}

// ---- kernel 5: fp32 WMMA GEMM  out = [mean | x] @ Bpack + bias -----------
// Block: 16 nodes x 256 cols, 8 waves; wave w owns col tiles w*32 and w*32+16.
// A panel double-buffered in LDS, filled with async global->LDS DMA.

__device__ __forceinline__ void mma_chunk(const float* __restrict__ A,  // LDS panel
                                          const float* __restrict__ Bpack,
                                          int k0, int row, int off,
                                          int colA, int colB,
                                          v8f& c0, v8f& c1) {
    const float* bp = Bpack + (size_t)(k0 >> 2) * (HH * 4);
    const size_t oA = (size_t)colA * 4 + off;
    const size_t oB = (size_t)colB * 4 + off;
    const float* ar = A + row * APAD + off;
    #pragma unroll
    for (int kk = 0; kk < 64; kk += 4) {
        v2f a  = *(const v2f*)(ar + kk);        // one ds_load_b64, conflict-free
        v2f b0 = *(const v2f*)(bp + oA);
        v2f b1 = *(const v2f*)(bp + oB);
        c0 = wmma_f32x4(a, b0, c0);
        c1 = wmma_f32x4(a, b1, c1);
        bp += HH * 4;           // advance one k-step group of 4
    }
}

__global__ __launch_bounds__(256) void k_gemm(
    const float* __restrict__ sums,             // (N, 2048) means
    const float* __restrict__ in_s, const float* __restrict__ in_a,
    const float* __restrict__ Bpack,            // (KTOT/4, 256, 4)
    const float* __restrict__ bias,             // (256,)
    float* __restrict__ out) {
    __shared__ float ldsA[2][16 * APAD];        // row-major A panels (16 x 64)

    const int m0   = blockIdx.x * 16;
    const int tid  = threadIdx.x;
    const int wid  = tid >> 5;
    const int lane = tid & 31;
    const int row  = lane & 15;                 // M index within tile
    const int off  = (lane < 16) ? 0 : 2;       // K sub-offset per A/B layout
    const int colA = wid * 32 + row;
    const int colB = colA + 16;

    // Cooperative DMA indices for the 16x64 A panel (16B per thread).
    const int lrow = tid >> 4;                  // 0..15 node row
    const int lkc  = (tid & 15) << 2;           // 0..60, float4 column

    const float* arow = sums + (size_t)(m0 + lrow) * KREL + lkc;
    const float* xrow = node_row(in_s, in_a, m0 + lrow) + lkc;

    // LDS byte addresses for this thread's 16B slot in each buffer
    // (low 32 bits of the flat shared pointer == LDS byte offset).
    const unsigned dst0 = (unsigned)(uintptr_t)&ldsA[0][lrow * APAD + lkc];
    const unsigned dst1 = (unsigned)(uintptr_t)&ldsA[1][lrow * APAD + lkc];

    v8f c0 = {};
    v8f c1 = {};

    // Prologue: kick off chunk 0.
    async_b128(dst0, arow);

    const int NCHUNK = KTOT / 64;               // 36
    for (int i = 0; i < NCHUNK; ++i) {
        wait_async0();                          // my DMA done...
        __syncthreads();                        // ...everyone's DMA done
        if (i + 1 < NCHUNK) {                   // prefetch next chunk into other buf
            int k1 = (i + 1) * 64;
            const float* g = (k1 < KREL) ? (arow + k1) : (xrow + (k1 - KREL));
            async_b128(((i + 1) & 1) ? dst1 : dst0, g);
        }
        const float* Acur = (i & 1) ? &ldsA[1][0] : &ldsA[0][0];
        mma_chunk(Acur, Bpack, i * 64, row, off, colA, colB, c0, c1);
    }

    // Epilogue: D tile layout — VGPR v: lanes 0-15 -> M=v, lanes 16-31 -> M=v+8.
    const float bA = bias[colA];
    const float bB = bias[colB];
    #pragma unroll
    for (int v = 0; v < 8; v++) {
        int mrow = v + ((lane < 16) ? 0 : 8);
        size_t o = out_off(m0 + mrow);
        out[o + colA] = c0[v] + bA;
        out[o + colB] = c1[v] + bB;
    }
}

// ---- host-side launcher ---------------------------------------------------
extern "C" void kernel_launch(void* const* d_in, const int* in_sizes, int n_in,
                              void* d_out, int out_size, void* d_ws, size_t ws_size,
                              hipStream_t stream) {
    const float*     in_s  = (const float*)d_in[0];
    const float*     in_a  = (const float*)d_in[1];
    const long long* eidx  = (const long long*)d_in[2];
    const long long* etype = (const long long*)d_in[3];
    const float*     w     = (const float*)d_in[4];
    const float*     root  = (const float*)d_in[5];
    const float*     bias  = (const float*)d_in[6];
    float*           out   = (float*)d_out;

    float*        sums  = (float*)d_ws;
    unsigned int* cnt   = (unsigned int*)((char*)d_ws + SUMS_ELEMS * sizeof(float));
    float*        Bpack = (float*)((char*)d_ws + (SUMS_ELEMS + CNT_ELEMS) * sizeof(float));

    // 1) zero sums + counts (contiguous region, multiple of 4 floats)
    size_t n4 = (SUMS_ELEMS + CNT_ELEMS) / 4;
    k_zero<<<(unsigned)((n4 + 255) / 256), 256, 0, stream>>>((float4*)d_ws, n4);

    // 2) pack B (independent of scatter; small)
    k_prepB<<<(unsigned)(BPACK_ELEMS / 256), 256, 0, stream>>>(w, root, Bpack);

    // 3) fused count + scatter-add: one block per edge
    k_scatter<<<EE, 256, 0, stream>>>(in_s, in_a, eidx, etype, sums, cnt);

    // 4) sums -> means
    k_scale<<<(unsigned)(SUMS_ELEMS / 4 / 256), 256, 0, stream>>>(sums, cnt);

    // 5) WMMA GEMM (+ root folded into K, + bias), write split (sent, act)
    k_gemm<<<NN / 16, 256, 0, stream>>>(sums, in_s, in_a, Bpack, bias, out);
}